// ODEFuncW_44074954391861
// MI455X (gfx1250) — compile-verified
//
#include <hip/hip_runtime.h>
#include <math.h>

#define NN 50000
#define DD 256
#define EE 400000
#define LPAD 258   // LDS row stride (floats): 16 rows, avoids 64-bank conflicts, keeps 8B align

typedef float v2f __attribute__((ext_vector_type(2)));
typedef float v8f __attribute__((ext_vector_type(8)));

// ---------------------------------------------------------------- zero ax
__global__ __launch_bounds__(256) void zero_kernel(float4* __restrict__ p) {
  size_t i = (size_t)blockIdx.x * 256 + threadIdx.x;
  p[i] = make_float4(0.f, 0.f, 0.f, 0.f);
}

// ------------------------------------------------- w_eff = (w * clip(d)) @ w^T
// 256 tiles of 16x16; 8 waves/block, 1 tile/wave, 32 blocks.
__global__ __launch_bounds__(256)
void weff_kernel(const float* __restrict__ w, const float* __restrict__ dvec,
                 float* __restrict__ weff) {
  __shared__ float dc[DD];
  const int tid = threadIdx.x;
  dc[tid] = fminf(fmaxf(dvec[tid], 0.f), 1.f);
  __syncthreads();

  const int wave = tid >> 5;
  const int lane = tid & 31;
  const int lm = lane & 15;
  const int lh = lane >> 4;                 // 0: K pair {k0,k0+1}, 1: {k0+2,k0+3}
  const int tile = blockIdx.x * 8 + wave;   // 0..255
  const int m0 = (tile >> 4) * 16;
  const int n0 = (tile & 15) * 16;

  v8f c = {};
  for (int k0 = 0; k0 < DD; k0 += 4) {
    const int ka = k0 + 2 * lh;
    v2f a = *(const v2f*)&w[(size_t)(m0 + lm) * DD + ka];   // A[m, ka..ka+1]
    a.x *= dc[ka];
    a.y *= dc[ka + 1];
    v2f b = *(const v2f*)&w[(size_t)(n0 + lm) * DD + ka];   // B[k,n] = w[n,k]
    c = __builtin_amdgcn_wmma_f32_16x16x4_f32(false, a, false, b,
                                              (short)0, c, false, false);
  }
#pragma unroll
  for (int r = 0; r < 8; ++r) {
    const int m = m0 + r + lh * 8;          // C layout: VGPR r -> M=r (lanes 0-15) / M=8+r
    weff[(size_t)m * DD + n0 + lm] = c[r];
  }
}

// --------------------------------------------------- per-row gate alph[n]
// one wave per row; 8 rows per 256-thread block; 6250 blocks.
__global__ __launch_bounds__(256)
void alpha_kernel(const float* __restrict__ x, const float* __restrict__ h,
                  const float* __restrict__ atr, const float* __restrict__ W_ih,
                  const float* __restrict__ W_hh, const float* __restrict__ b_ih,
                  const float* __restrict__ b_hh, float* __restrict__ alph) {
  const int n = blockIdx.x * 8 + (threadIdx.x >> 5);
  const int lane = threadIdx.x & 31;
  const float* xr = x + (size_t)n * DD;

  float s0 = 0.f, s1 = 0.f;
#pragma unroll
  for (int j = 0; j < 8; ++j) {
    const int col = lane + 32 * j;
    const float xv = xr[col];
    s0 = fmaf(xv, W_ih[col], s0);
    s1 = fmaf(xv, W_ih[DD + col], s1);
  }
#pragma unroll
  for (int off = 16; off > 0; off >>= 1) {
    s0 += __shfl_xor(s0, off);
    s1 += __shfl_xor(s1, off);
  }
  if (lane == 0) {
    const float h0 = h[2 * n], h1 = h[2 * n + 1];
    const float a0 = tanhf(s0 + b_ih[0] + h0 * W_hh[0] + h1 * W_hh[1] + b_hh[0]);
    const float a1 = tanhf(s1 + b_ih[1] + h0 * W_hh[2] + h1 * W_hh[3] + b_hh[1]);
    const float at = atr[n] * a0 + a1;
    alph[n] = 1.f / (1.f + expf(-at));
  }
}

// -------------------------------------------- SpMM: ax[row] += val * x[col]
// one wave per edge; 8 edges per block; 50000 blocks. Coalesced atomics.
__global__ __launch_bounds__(256)
void spmm_kernel(const int* __restrict__ erow, const int* __restrict__ ecol,
                 const float* __restrict__ ev, const float* __restrict__ x,
                 float* __restrict__ ax) {
  const int e = blockIdx.x * 8 + (threadIdx.x >> 5);
  const int lane = threadIdx.x & 31;
  const int r = erow[e];
  const int c = ecol[e];
  const float v = ev[e];
  const float* xr = x + (size_t)c * DD;
  float* axr = ax + (size_t)r * DD;
#pragma unroll
  for (int j = 0; j < 8; ++j) {
    const int col = lane + 32 * j;
    atomicAdd(axr + col, v * xr[col]);
  }
}

// ---------------------- fused: xw = x @ w_eff ; f = alph*0.5*(ax-x)+xw-x+x0
// one block per 16-row strip (3125 blocks); 8 waves; each wave owns two
// 16x16 output tiles. x-strip staged in padded LDS for A fragments.
__global__ __launch_bounds__(256)
void fused_kernel(const float* __restrict__ x, const float* __restrict__ x0,
                  const float* __restrict__ ax, const float* __restrict__ alph,
                  const float* __restrict__ weff, float* __restrict__ out) {
  __shared__ float lx[16 * LPAD];
  const int row0 = blockIdx.x * 16;
  const int tid = threadIdx.x;

  // cooperative load of 16x256 x-strip (1024 float4, 4 per thread)
  {
    const float4* xs = (const float4*)(x + (size_t)row0 * DD);
#pragma unroll
    for (int i = 0; i < 4; ++i) {
      const int idx = tid + i * 256;        // 0..1023
      const int r = idx >> 6;               // 64 float4 per row
      const int c = idx & 63;
      const float4 v = xs[r * 64 + c];
      float* dst = &lx[r * LPAD + c * 4];
      dst[0] = v.x; dst[1] = v.y; dst[2] = v.z; dst[3] = v.w;
    }
  }
  __syncthreads();

  const int wave = tid >> 5;
  const int lane = tid & 31;
  const int lm = lane & 15;
  const int lh = lane >> 4;
  const int n0 = wave * 32;                 // this wave: cols [n0,n0+16) and [n0+16,n0+32)

  v8f c0 = {};
  v8f c1 = {};
  for (int k0 = 0; k0 < DD; k0 += 4) {
    const int ka = k0 + 2 * lh;
    const v2f a = *(const v2f*)&lx[lm * LPAD + ka];        // A frag from LDS
    const float* wrow0 = weff + (size_t)ka * DD;           // B rows ka, ka+1
    const float* wrow1 = wrow0 + DD;
    if (k0 + 4 < DD)
      __builtin_prefetch(wrow0 + 4 * DD + n0 + lm, 0, 3);  // next K rows (L2-resident)
    v2f b0, b1;
    b0.x = wrow0[n0 + lm];       b0.y = wrow1[n0 + lm];
    b1.x = wrow0[n0 + 16 + lm];  b1.y = wrow1[n0 + 16 + lm];
    c0 = __builtin_amdgcn_wmma_f32_16x16x4_f32(false, a, false, b0,
                                               (short)0, c0, false, false);
    c1 = __builtin_amdgcn_wmma_f32_16x16x4_f32(false, a, false, b1,
                                               (short)0, c1, false, false);
  }

  // epilogue: C layout VGPR r -> M = r + 8*lh, N = lm
#pragma unroll
  for (int r = 0; r < 8; ++r) {
    const int m = r + lh * 8;
    const size_t row = (size_t)(row0 + m);
    const float al = alph[row] * 0.5f;
    {
      const size_t off = row * DD + n0 + lm;
      const float xv = lx[m * LPAD + n0 + lm];
      out[off] = al * (ax[off] - xv) + c0[r] - xv + x0[off];
    }
    {
      const size_t off = row * DD + n0 + 16 + lm;
      const float xv = lx[m * LPAD + n0 + 16 + lm];
      out[off] = al * (ax[off] - xv) + c1[r] - xv + x0[off];
    }
  }
}

// ----------------------------------------------------------------- launcher
extern "C" void kernel_launch(void* const* d_in, const int* in_sizes, int n_in,
                              void* d_out, int out_size, void* d_ws, size_t ws_size,
                              hipStream_t stream) {
  (void)in_sizes; (void)n_in; (void)out_size; (void)ws_size;
  // setup_inputs order: t, x, x0, edge_rows, edge_cols, edge_vals, h,
  //                     alpha_train, W_ih, W_hh, b_ih, b_hh, w, d
  const float* x    = (const float*)d_in[1];
  const float* x0   = (const float*)d_in[2];
  const int*   erow = (const int*)d_in[3];
  const int*   ecol = (const int*)d_in[4];
  const float* ev   = (const float*)d_in[5];
  const float* h    = (const float*)d_in[6];
  const float* atr  = (const float*)d_in[7];
  const float* W_ih = (const float*)d_in[8];
  const float* W_hh = (const float*)d_in[9];
  const float* b_ih = (const float*)d_in[10];
  const float* b_hh = (const float*)d_in[11];
  const float* w    = (const float*)d_in[12];
  const float* dv   = (const float*)d_in[13];
  float* out = (float*)d_out;

  // workspace layout (floats): [weff 64K][alph 50176][ax N*D]  (~51.7 MB)
  float* weff = (float*)d_ws;
  float* alph = weff + (size_t)DD * DD;
  float* ax   = alph + 50176;              // N rounded up; keeps 256B alignment

  // 1) clear accumulator (N*D/4 float4 = 3.2M; 12500 blocks)
  zero_kernel<<<(NN * DD) / (256 * 4), 256, 0, stream>>>((float4*)ax);
  // 2) w_eff GEMM (256 tiles / 8 per block)
  weff_kernel<<<32, 256, 0, stream>>>(w, dv, weff);
  // 3) per-row gate (8 rows per block)
  alpha_kernel<<<NN / 8, 256, 0, stream>>>(x, h, atr, W_ih, W_hh, b_ih, b_hh, alph);
  // 4) SpMM scatter-add (8 edges per block)
  spmm_kernel<<<EE / 8, 256, 0, stream>>>(erow, ecol, ev, x, ax);
  // 5) fused main GEMM + epilogue (16 rows per block)
  fused_kernel<<<NN / 16, 256, 0, stream>>>(x, x0, ax, alph, weff, out);
}